// AugmentationModule_16140487098637
// MI455X (gfx1250) — compile-verified
//
#include <hip/hip_runtime.h>
#include <math.h>

typedef __attribute__((ext_vector_type(2))) float v2f;
typedef __attribute__((ext_vector_type(8))) float v8f;

#define NTOT   16384
#define NGRAPH 256
#define KNN    50
#define NBINS  5
#define BIGF   1.0e10f   // reference's masked-distance sentinel
#define SENT   3.0e38f   // internal "taken / masked" sentinel (> BIGF)
#define MAXM   512       // max nodes per graph handled (mean 64, max ~100)

// One wave (32 threads) per graph.
// - Graph coords (x,y,z,|p|^2) staged once into LDS.
// - 16x16 d^2 tiles via V_WMMA_F32_16X16X4_F32 with the row-norm fused in
//   through the C operand:  A row i = (-2x,-2y,-2z,1), B col j = (x,y,z,|p_j|^2),
//   C[M][*] = |p_M|^2  =>  D = |p_i|^2 - 2 p_i.p_j + |p_j|^2 = d2.
// - Two row-tiles (32 rows) per sweep so the K-selection phase uses all 32 lanes.
__global__ __launch_bounds__(32)
void knn_rdf_wmma_kernel(const float* __restrict__ pos,
                         const int*   __restrict__ batch,
                         float*       __restrict__ out) {
  __shared__ float px[MAXM], py[MAXM], pz[MAXM], psq[MAXM];
  __shared__ float d2row[32 * MAXM];   // 32 rows x up-to-MAXM cols of d2

  const int g    = blockIdx.x;
  const int lane = threadIdx.x;        // wave32
  const int l15  = lane & 15;

  // lower_bound(batch, g) / lower_bound(batch, g+1): graph range [s, e)
  int s, e;
  {
    int lo = 0, hi = NTOT;
    while (lo < hi) { int mid = (lo + hi) >> 1; if (batch[mid] <  g)     lo = mid + 1; else hi = mid; }
    s = lo;
    lo = s; hi = NTOT;
    while (lo < hi) { int mid = (lo + hi) >> 1; if (batch[mid] <  g + 1) lo = mid + 1; else hi = mid; }
    e = lo;
  }
  int m = e - s;
  if (m <= 0) return;
  if (m > MAXM) m = MAXM;              // safety clamp (prob ~0 for this dist)

  const int mPad      = (m + 15) & ~15;
  const int ncolTiles = mPad >> 4;

  // ---- stage this graph's coordinates into LDS (single global pass) ----
  for (int j = lane; j < mPad; j += 32) {
    float x = 0.f, y = 0.f, z = 0.f, q = 0.f;
    if (j < m) {
      const int idx = s + j;
      x = pos[3 * idx + 0];
      y = pos[3 * idx + 1];
      z = pos[3 * idx + 2];
      q = x * x + y * y + z * z;
    }
    px[j] = x; py[j] = y; pz[j] = z; psq[j] = q;
  }
  __syncthreads();

  const long long NK = (long long)NTOT * KNN;
  float* out_src  = out;
  float* out_dst  = out + NK;
  float* out_dist = out + 2 * NK;
  float* out_rdf  = out + 3 * NK;

  for (int rowBase = 0; rowBase < m; rowBase += 32) {
    // ---- A fragments for the two 16-row tiles ----
    const int  r0  = rowBase + l15;        const bool v0r = r0 < m;
    const int  r1  = rowBase + 16 + l15;   const bool v1r = r1 < m;
    const int  rr0 = v0r ? r0 : 0;
    const int  rr1 = v1r ? r1 : 0;
    v2f a0, a1;
    if (lane < 16) {
      a0.x = -2.0f * px[rr0]; a0.y = -2.0f * py[rr0];            // K=0,1
      a1.x = -2.0f * px[rr1]; a1.y = -2.0f * py[rr1];
    } else {
      a0.x = -2.0f * pz[rr0]; a0.y = v0r ? 1.0f : 0.0f;          // K=2,3
      a1.x = -2.0f * pz[rr1]; a1.y = v1r ? 1.0f : 0.0f;
    }

    // ---- preload C with the row norms (fused into D = A*B + C) ----
    v8f c0, c1;
    #pragma unroll
    for (int v = 0; v < 8; ++v) {
      const int rL = (lane < 16) ? v : (v + 8);
      int i0 = rowBase + rL;       if (i0 >= MAXM) i0 = MAXM - 1;
      int i1 = rowBase + 16 + rL;  if (i1 >= MAXM) i1 = MAXM - 1;
      c0[v] = psq[i0];
      c1[v] = psq[i1];
    }

    // ---- sweep column tiles: two WMMAs per tile, masked store to LDS ----
    for (int ct = 0; ct < ncolTiles; ++ct) {
      const int colBase = ct << 4;
      const int cloc    = colBase + l15;
      v2f b;
      if (lane < 16) { b.x = px[cloc]; b.y = py[cloc]; }         // K=0,1
      else           { b.x = pz[cloc]; b.y = psq[cloc]; }        // K=2,3

      // (neg_a, A, neg_b, B, c_mod, C, reuse_a, reuse_b)
      v8f d0 = __builtin_amdgcn_wmma_f32_16x16x4_f32(
          false, a0, false, b, (short)0, c0, false, false);
      v8f d1 = __builtin_amdgcn_wmma_f32_16x16x4_f32(
          false, a1, false, b, (short)0, c1, false, false);

      // D layout: lane<16 -> (M=v, N=lane), lane>=16 -> (M=v+8, N=lane-16)
      #pragma unroll
      for (int v = 0; v < 8; ++v) {
        const int  rL   = (lane < 16) ? v : (v + 8);
        const int  row0 = rowBase + rL;
        const int  row1 = rowBase + 16 + rL;
        const bool m0   = (cloc >= m) || (cloc == row0);
        const bool m1   = (cloc >= m) || (cloc == row1);
        d2row[rL * MAXM + cloc]        = m0 ? SENT : d0[v];
        d2row[(rL + 16) * MAXM + cloc] = m1 ? SENT : d1[v];
      }
    }
    __syncthreads();

    // ---- per-row K-smallest selection: every lane owns one row ----
    const int myRow = rowBase + lane;
    if (myRow < m) {
      float* row = &d2row[lane * MAXM];
      const int       gi    = s + myRow;
      const long long obase = (long long)gi * KNN;
      int pcount = 0;  // padding counter for graphs with < K valid neighbors

      for (int k = 0; k < KNN; ++k) {
        float best = SENT; int bestj = -1;
        for (int j = 0; j < m; ++j) {        // ascending scan => lowest-index tie-break
          const float v = row[j];
          if (v < best) { best = v; bestj = j; }
        }
        float d2v; int srcIdx;
        if (bestj >= 0 && best < 1.0e9f) {   // real within-graph neighbor
          row[bestj] = SENT;
          d2v = best; srcIdx = s + bestj;
        } else {
          // JAX top_k tie order among 1e10-masked cols: 0..s-1, i, e, e+1, ...
          int padj;
          if      (pcount <  s) padj = pcount;
          else if (pcount == s) padj = gi;
          else                  padj = e + (pcount - s - 1);
          pcount++;
          d2v = BIGF; srcIdx = padj;
        }
        const float dist = sqrtf(fmaxf(d2v, 1e-12f));
        out_src[obase + k]  = (float)srcIdx;
        out_dst[obase + k]  = (float)gi;
        out_dist[obase + k] = dist;
        // Gaussian RDF: centers 0,2.5,5,7.5,10 ; gamma = 1/(2*2.5^2) = 0.08
        #pragma unroll
        for (int b = 0; b < NBINS; ++b) {
          const float t = dist - 2.5f * (float)b;
          out_rdf[(obase + k) * NBINS + b] = __expf(-0.08f * t * t);
        }
      }
    }
    __syncthreads();
  }
}

extern "C" void kernel_launch(void* const* d_in, const int* in_sizes, int n_in,
                              void* d_out, int out_size, void* d_ws, size_t ws_size,
                              hipStream_t stream) {
  const float* pos   = (const float*)d_in[0]; // [N,3] f32
  const int*   batch = (const int*)d_in[1];   // [N] sorted graph ids
  float*       out   = (float*)d_out;         // src|dst|dist|rdf flat
  (void)in_sizes; (void)n_in; (void)out_size; (void)d_ws; (void)ws_size;

  knn_rdf_wmma_kernel<<<NGRAPH, 32, 0, stream>>>(pos, batch, out);
}